// Net_88493506167258
// MI455X (gfx1250) — compile-verified
//
#include <hip/hip_runtime.h>
#include <stdint.h>

typedef __attribute__((ext_vector_type(2))) float    v2f;
typedef __attribute__((ext_vector_type(8))) float    v8f;
typedef __attribute__((ext_vector_type(4))) unsigned v4u;
typedef __attribute__((ext_vector_type(8))) unsigned v8u;

#define IMG_W 4096
#define TILE  64      // output tile (64x64)
#define IW    72      // img halo tile:   [y0-4 .. y0+67] x [x0-4 .. x0+67]
#define HBW   68      // hblur width:     cols x0-2 .. x0+65
#define HBH   72      // hblur height:    rows y0-4 .. y0+67
#define BW    68      // blurred:         rows/cols (y0-2.., x0-2..)
#define GW    66      // gradient region: rows/cols (y0-1.., x0-1..)
#define NT    512     // threads per block (16 waves)

__global__ __launch_bounds__(NT)
void canny_fused(const float* __restrict__ img,
                 const float* __restrict__ gauss,
                 const float* __restrict__ sobel,
                 const float* __restrict__ dirw,
                 float* __restrict__ out)
{
    __shared__ float s_img[3 * IW * IW];   // channel-major planes; plane 0 reused as s_blur
    __shared__ float s_hb[HBH * HBW];
    __shared__ float s_mag[GW * GW];
    __shared__ float s_gx[GW * GW];
    __shared__ float s_gy[GW * GW];
    __shared__ float s_g8[8];              // padded gaussian taps for WMMA A-matrix

    float* s_blur = s_img;                 // 68*68 = 4624 <= 5184 (one img plane)

    const int tid = threadIdx.x;
    const int x0 = blockIdx.x * TILE;
    const int y0 = blockIdx.y * TILE;

    // halo fully inside image? (block-uniform)
    const bool interior = (x0 >= 4) && (y0 >= 4) &&
                          (x0 <= IMG_W - BW) && (y0 <= IMG_W - BW); // x0+67 <= 4095
    const bool wave0 = (tid >> 5) == 0;

    // ---- Stage 72x72x3 halo tile into LDS: issue TDM first, wait late --------
    if (interior && wave0) {
        const uint64_t gaddr = (uint64_t)(uintptr_t)img +
            ((uint64_t)(y0 - 4) * IMG_W + (uint64_t)(x0 - 4)) * 4ull;
        const uint32_t lds = (uint32_t)(uintptr_t)(void*)s_img;
        const uint64_t planeStride = (uint64_t)IMG_W * IMG_W; // dim1 stride (elements)

        // D# group 0: count=1 | lds_addr | global_addr | type=2
        v4u g0 = { 1u,
                   lds,
                   (uint32_t)gaddr,
                   (((uint32_t)(gaddr >> 32)) & 0x01FFFFFFu) | 0x80000000u };
        // D# group 1: data_size=4B, dims/strides, tile 72x72x3
        v8u g1 = { (2u << 16),                                       // data_size=2 (4B)
                   ((uint32_t)IMG_W & 0xFFFFu) << 16,                 // tensor_dim0 lo16
                   ((uint32_t)IMG_W >> 16) | (((uint32_t)IMG_W & 0xFFFFu) << 16),
                   ((uint32_t)IMG_W >> 16) | ((uint32_t)IW << 16),    // tile_dim0=72
                   (uint32_t)IW | (3u << 16),                         // tile_dim1=72, tile_dim2=3
                   (uint32_t)IMG_W,                                   // tensor_dim0_stride lo32
                   (uint32_t)((planeStride & 0xFFFFull) << 16),       // dim0 hi16 | dim1 lo16
                   (uint32_t)(planeStride >> 16) };                   // tensor_dim1_stride hi32
        // D# group 2: tensor_dim2=3, rest 0 ; group 3: zeros
        v4u g2 = { 3u, 0u, 0u, 0u };
        v4u g3 = { 0u, 0u, 0u, 0u };

        asm volatile("tensor_load_to_lds %0, %1, %2, %3"
                     :: "s"(g0), "s"(g1), "s"(g2), "s"(g3) : "memory");
    }

    if (tid < 8) s_g8[tid] = (tid < 5) ? gauss[tid] : 0.0f;

    if (!interior) {
        // border tiles: guarded loads with zero fill (matches zero padding)
        for (int i = tid; i < 3 * IW * IW; i += NT) {
            int c = i / (IW * IW), r = i - c * (IW * IW);
            int ty = r / IW, tx = r - ty * IW;
            int ay = y0 - 4 + ty, ax = x0 - 4 + tx;
            float v = 0.f;
            if ((unsigned)ay < (unsigned)IMG_W && (unsigned)ax < (unsigned)IMG_W)
                v = img[(size_t)c * IMG_W * IMG_W + (size_t)ay * IMG_W + ax];
            s_img[i] = v;
        }
    }
    if (interior && wave0) __builtin_amdgcn_s_wait_tensorcnt(0);
    __syncthreads();

    // per-thread filter weights (stay in VGPRs)
    const float G0 = gauss[0], G1 = gauss[1], G2 = gauss[2], G3 = gauss[3], G4 = gauss[4];
    float SW[9];
#pragma unroll
    for (int i = 0; i < 9; ++i) SW[i] = sobel[i];

    // 2D work decomposition: one row x 32 consecutive cols per wave (no divides,
    // conflict-free LDS rows)
    const int rB = tid >> 5;   // row lane group: 0..15, stride 16
    const int cB = tid & 31;   // col within row: stride 32

    // ---- Per-channel: hblur (VALU) -> vblur (WMMA) -> sobel+mag accumulate ----
#pragma unroll
    for (int c = 0; c < 3; ++c) {
        const float* ip = &s_img[c * IW * IW];

        // horizontal 5-tap gaussian: 72 rows x 68 cols
        for (int hr = rB; hr < HBH; hr += 16)
            for (int hc = cB; hc < HBW; hc += 32) {
                const float* p = ip + hr * IW + hc;
                s_hb[hr * HBW + hc] =
                    G0 * p[0] + G1 * p[1] + G2 * p[2] + G3 * p[3] + G4 * p[4];
            }
        __syncthreads();

        // vertical 5-tap gaussian as banded matmul on the matrix pipe:
        // blur[brB+M][bcB+N] = sum_K Gband[M][K] * hblur[brB+K][bcB+N], K=0..19
        {
            const int wave = tid >> 5;
            const int lane = tid & 31;
            const int half = lane >> 4;  // lanes 16-31 hold K+2 (A,B) / M+8 (D)
            const int ln   = lane & 15;

            for (int t = wave; t < 25; t += 16) {
                int tr = t / 5, tc = t - tr * 5;
                int brB = tr * 16; if (brB > 52) brB = 52;  // overlap last tile: 68 rows
                int bcB = tc * 16; if (bcB > 52) bcB = 52;  // overlap last tile: 68 cols
                v8f acc = {};
#pragma unroll
                for (int k4 = 0; k4 < 5; ++k4) {
                    const int k0 = 4 * k4 + 2 * half;
                    // A[M][K] = g[K-M] (banded), via zero-padded LDS gauss table
                    int u0 = k0 - ln, u1 = k0 + 1 - ln;
                    unsigned i0 = ((unsigned)u0 > 7u) ? 7u : (unsigned)u0;
                    unsigned i1 = ((unsigned)u1 > 7u) ? 7u : (unsigned)u1;
                    v2f a; a.x = s_g8[i0]; a.y = s_g8[i1];
                    // B[K][N] = hblur rows
                    v2f b;
                    b.x = s_hb[(brB + k0) * HBW + bcB + ln];
                    b.y = s_hb[(brB + k0 + 1) * HBW + bcB + ln];
                    acc = __builtin_amdgcn_wmma_f32_16x16x4_f32(
                              false, a, false, b, (short)0, acc, false, false);
                }
#pragma unroll
                for (int r = 0; r < 8; ++r)
                    s_blur[(brB + r + 8 * half) * BW + bcB + ln] = acc[r];
            }
        }
        __syncthreads();

        // border tiles: reference zero-pads 'blurred' before sobel
        if (!interior) {
            for (int i = tid; i < BW * BW; i += NT) {
                int br = i / BW, bc = i - br * BW;
                int ay = y0 - 2 + br, ax = x0 - 2 + bc;
                if ((unsigned)ay >= (unsigned)IMG_W || (unsigned)ax >= (unsigned)IMG_W)
                    s_blur[i] = 0.f;
            }
            __syncthreads();
        }

        // sobel + magnitude accumulation on 66x66 gradient region
        for (int gr = rB; gr < GW; gr += 16)
            for (int gc = cB; gc < GW; gc += 32) {
                const float* bp = &s_blur[gr * BW + gc];
                float b00 = bp[0],        b01 = bp[1],          b02 = bp[2];
                float b10 = bp[BW],       b11 = bp[BW + 1],     b12 = bp[BW + 2];
                float b20 = bp[2 * BW],   b21 = bp[2 * BW + 1], b22 = bp[2 * BW + 2];
                float gx = SW[0]*b00 + SW[1]*b01 + SW[2]*b02
                         + SW[3]*b10 + SW[4]*b11 + SW[5]*b12
                         + SW[6]*b20 + SW[7]*b21 + SW[8]*b22;
                float gy = SW[0]*b00 + SW[3]*b01 + SW[6]*b02
                         + SW[1]*b10 + SW[4]*b11 + SW[7]*b12
                         + SW[2]*b20 + SW[5]*b21 + SW[8]*b22;
                float m = sqrtf(gx * gx + gy * gy + 1e-8f);
                int idx = gr * GW + gc;
                if (c == 0) {            // channel loop is unrolled: static branch
                    s_mag[idx] = m;  s_gx[idx] = gx;  s_gy[idx] = gy;
                } else {
                    s_mag[idx] += m; s_gx[idx] += gx; s_gy[idx] += gy;
                }
            }
        __syncthreads();
    }

    // border tiles: reference zero-pads grad_mag before directional filters
    if (!interior) {
        for (int i = tid; i < GW * GW; i += NT) {
            int gr = i / GW, gc = i - gr * GW;
            int ay = y0 - 1 + gr, ax = x0 - 1 + gc;
            if ((unsigned)ay >= (unsigned)IMG_W || (unsigned)ax >= (unsigned)IMG_W)
                s_mag[i] = 0.f;
        }
        __syncthreads();
    }

    // ---- orientation, NMS, double threshold ----------------------------------
    const float K180 = 180.0f / 3.14159f;   // reference uses 3.14159, not pi
    for (int op = tid; op < TILE * TILE; op += NT) {
        int oy = op >> 6, ox = op & 63;
        int cy = oy + 1, cx = ox + 1;
        float mag = s_mag[cy * GW + cx];
        float vx  = s_gx[cy * GW + cx];
        float vy  = s_gy[cy * GW + cx];
        float ori = atan2f(vy, vx) * K180 + 180.0f;
        float sct = rintf(ori * (1.0f / 45.0f));        // jnp.round = RNE
        int kp = (((int)sct) + 8) & 7;
        int kn = (kp + 4) & 7;
        // dir_w[k] = center - neighbor(dr,dc); nibble-packed offset tables (+1 biased)
        int drp = (int)((0x00012221u >> (4 * kp)) & 7u) - 1;
        int dcp = (int)((0x21000122u >> (4 * kp)) & 7u) - 1;
        int drn = (int)((0x00012221u >> (4 * kn)) & 7u) - 1;
        int dcn = (int)((0x21000122u >> (4 * kn)) & 7u) - 1;
        float csp = mag - s_mag[(cy + drp) * GW + cx + dcp];
        float csn = mag - s_mag[(cy + drn) * GW + cx + dcn];
        bool is_max = fminf(csp, csn) > 0.0f;
        bool keep = is_max && (mag >= 6.0f) && (mag <= 50.0f);
        out[(size_t)(y0 + oy) * IMG_W + (x0 + ox)] = keep ? 1.0f : 0.0f;
    }
}

extern "C" void kernel_launch(void* const* d_in, const int* in_sizes, int n_in,
                              void* d_out, int out_size, void* d_ws, size_t ws_size,
                              hipStream_t stream) {
    (void)in_sizes; (void)n_in; (void)out_size; (void)d_ws; (void)ws_size;
    const float* img   = (const float*)d_in[0];
    const float* gauss = (const float*)d_in[1];
    const float* sobel = (const float*)d_in[2];
    const float* dirw  = (const float*)d_in[3];
    dim3 grid(IMG_W / TILE, IMG_W / TILE);
    canny_fused<<<grid, dim3(NT), 0, stream>>>(img, gauss, sobel, dirw, (float*)d_out);
}